// GraphEmbedder_46265387712832
// MI455X (gfx1250) — compile-verified
//
#include <hip/hip_runtime.h>
#include <math.h>

// Problem constants (from reference)
#define Nn   51200      // total nodes
#define Bg   256        // graphs
#define NPGc 200        // nodes per graph
#define Eg   819200     // edges
#define Hh   256        // hidden

typedef float v8f __attribute__((ext_vector_type(8)));
typedef float v2f __attribute__((ext_vector_type(2)));

// ---------------------------------------------------------------------------
// WMMA fp32 GEMM:  C[M x Nc] = A[M x K] * W[K x Nc]   (optionally += existing C)
// Block tile 64x128, K-chunk 32. 8 waves (wave32): 4 in M, 2 in N.
// Each wave: 16x64 strip = 4x v_wmma_f32_16x16x4_f32 accumulators.
// W tile kept TRANSPOSED in LDS so each B fragment (K-pair) is a single
// 8B-aligned ds_load_b64 directly into the WMMA operand register pair.
// Requires M%64==0, Nc%128==0, K%32==0 (true for all GEMMs here).
// ---------------------------------------------------------------------------
#define TM 64
#define TN 128
#define TK 32
#define PAD 4           // row stride 36: 16B-aligned rows, conflict-free frags

__global__ __launch_bounds__(256)
void wmma_gemm_f32(const float* __restrict__ A, int lda,
                   const float* __restrict__ W, int ldw,
                   float* __restrict__ C, int ldc,
                   int K, int accum)
{
    __shared__ float As[TM][TK + PAD];   // row-major A tile
    __shared__ float Ws[TN][TK + PAD];   // TRANSPOSED W tile: Ws[col][k]

    const int bm   = blockIdx.y * TM;
    const int bn   = blockIdx.x * TN;
    const int tid  = threadIdx.x;
    const int lane = tid & 31;
    const int wave = tid >> 5;
    const int half = lane >> 4;        // lane half selects K-pair (A/B) or M+8 (C)
    const int lid  = lane & 15;
    const int wm   = (wave & 3) * 16;  // wave row offset in block tile
    const int wn   = (wave >> 2) * 64; // wave col offset in block tile

    v8f acc[4];
    if (accum) {
        #pragma unroll
        for (int t = 0; t < 4; ++t)
            #pragma unroll
            for (int r = 0; r < 8; ++r)
                acc[t][r] = C[(size_t)(bm + wm + r + 8 * half) * ldc + (bn + wn + t * 16 + lid)];
    } else {
        #pragma unroll
        for (int t = 0; t < 4; ++t)
            #pragma unroll
            for (int r = 0; r < 8; ++r)
                acc[t][r] = 0.0f;
    }

    for (int k0 = 0; k0 < K; k0 += TK) {
        // A tile 64x32: 512 float4, b128 global loads + b128 LDS stores
        #pragma unroll
        for (int j = tid; j < TM * TK / 4; j += 256) {
            int flat = j << 2;
            int r = flat >> 5, c = flat & (TK - 1);
            float4 v = *(const float4*)&A[(size_t)(bm + r) * lda + (k0 + c)];
            *(float4*)&As[r][c] = v;
        }
        // W tile 32x128, stored transposed: b128 global load, 4 scalar LDS stores
        #pragma unroll
        for (int j = tid; j < TK * TN / 4; j += 256) {
            int flat = j << 2;
            int r = flat >> 7, c = flat & (TN - 1);
            float4 v = *(const float4*)&W[(size_t)(k0 + r) * ldw + (bn + c)];
            Ws[c + 0][r] = v.x;
            Ws[c + 1][r] = v.y;
            Ws[c + 2][r] = v.z;
            Ws[c + 3][r] = v.w;
        }
        __syncthreads();
        #pragma unroll
        for (int kk = 0; kk < TK; kk += 4) {
            // A fragment 16x4: lanes 0-15 hold K=kk,kk+1; lanes 16-31 K=kk+2,kk+3
            v2f a = *(const v2f*)&As[wm + lid][kk + 2 * half];
            #pragma unroll
            for (int t = 0; t < 4; ++t) {
                // B fragment 4x16: consecutive K-pair from transposed tile
                v2f b = *(const v2f*)&Ws[wn + t * 16 + lid][kk + 2 * half];
                acc[t] = __builtin_amdgcn_wmma_f32_16x16x4_f32(
                    false, a, false, b, (short)0, acc[t], false, false);
            }
        }
        __syncthreads();
    }

    #pragma unroll
    for (int t = 0; t < 4; ++t)
        #pragma unroll
        for (int r = 0; r < 8; ++r)
            C[(size_t)(bm + wm + r + 8 * half) * ldc + (bn + wn + t * 16 + lid)] = acc[t][r];
}

// ---------------------------------------------------------------------------
// Attention pool: one block (256 threads) per graph.
// pooled[b,j] = sum_p softmax_j(logits[n]+ba) * (feat[n,j]+bf[j]),  n = b*200+p
// Writes P[b, 0:256] = pooled, P[b, 256:512] = g (concat for the Wg GEMM).
// ---------------------------------------------------------------------------
__global__ __launch_bounds__(256)
void pool_attn(const float* __restrict__ T1, const float* __restrict__ T2,
               const float* __restrict__ ba, const float* __restrict__ bf,
               const float* __restrict__ G,  float* __restrict__ P)
{
    const int b   = blockIdx.x;
    const int tid = threadIdx.x;
    const int wid = tid >> 5;
    __shared__ float red[8];

    const float bav = ba[tid];
    const float bfv = bf[tid];
    float acc = 0.0f;
    const int base = b * NPGc;

    for (int p = 0; p < NPGc; ++p) {
        const size_t n = (size_t)(base + p);
        float l = T1[n * Hh + tid] + bav;
        // block max over 256 channels
        float m = l;
        #pragma unroll
        for (int s = 16; s > 0; s >>= 1) m = fmaxf(m, __shfl_xor(m, s, 32));
        if ((tid & 31) == 0) red[wid] = m;
        __syncthreads();
        float bm = red[0];
        #pragma unroll
        for (int w = 1; w < 8; ++w) bm = fmaxf(bm, red[w]);
        __syncthreads();
        // block sum of exp
        float e = expf(l - bm);
        float s = e;
        #pragma unroll
        for (int o = 16; o > 0; o >>= 1) s += __shfl_xor(s, o, 32);
        if ((tid & 31) == 0) red[wid] = s;
        __syncthreads();
        float bs = 0.0f;
        #pragma unroll
        for (int w = 0; w < 8; ++w) bs += red[w];
        __syncthreads();

        acc += (e / bs) * (T2[n * Hh + tid] + bfv);
    }
    P[(size_t)b * 512 + tid]       = acc;
    P[(size_t)b * 512 + Hh + tid]  = G[(size_t)b * Hh + tid];
}

// G += tanh(GT + bg);  grid Bg, block 256
__global__ __launch_bounds__(256)
void tanh_add(const float* __restrict__ GT, const float* __restrict__ bg,
              float* __restrict__ G)
{
    size_t i = (size_t)blockIdx.x * Hh + threadIdx.x;
    G[i] += tanhf(GT[i] + bg[threadIdx.x]);
}

// Build x_cat and pre-fill agg. agg[:,0:d]=0 (scatter target); agg[:,d:]=deg*attn
// (exact because edges are always within-graph).  grid Nn, block 256. float4.
__global__ __launch_bounds__(256)
void prep_sage(const float* __restrict__ X, const float* __restrict__ G,
               const float* __restrict__ DEG,
               float* __restrict__ XCAT, float* __restrict__ AGG, int d)
{
    const int n   = blockIdx.x;
    const int g   = n / NPGc;
    const float deg = DEG[n];
    const int ldc4 = (d + Hh) >> 2;
    const int d4   = d >> 2;
    float4*       XC4 = (float4*)XCAT + (size_t)n * ldc4;
    float4*       AG4 = (float4*)AGG  + (size_t)n * ldc4;
    const float4* X4  = (const float4*)X + (size_t)n * d4;
    const float4* G4  = (const float4*)G + (size_t)g * (Hh >> 2);
    const float4 z4 = make_float4(0.f, 0.f, 0.f, 0.f);
    for (int j = threadIdx.x; j < d4; j += 256) {
        XC4[j] = X4[j];
        AG4[j] = z4;
    }
    for (int j = threadIdx.x; j < (Hh >> 2); j += 256) {
        float4 gv = G4[j];
        XC4[d4 + j] = gv;
        AG4[d4 + j] = make_float4(deg * gv.x, deg * gv.y, deg * gv.z, deg * gv.w);
    }
}

// Edge scatter for the x-columns only: one block per edge, d threads.
__global__ void scatter_add(const int* __restrict__ ei, const float* __restrict__ X,
                            float* __restrict__ AGG, int d, int ldagg)
{
    const int e    = blockIdx.x;
    const int srcn = ei[e];
    const int dstn = ei[Eg + e];
    const int j    = threadIdx.x;
    atomicAdd(&AGG[(size_t)dstn * ldagg + j], X[(size_t)srcn * d + j]);
}

// Y = relu(Y + bl);  grid Nn, block dout/4, float4
__global__ void bias_relu(float4* __restrict__ Y, const float4* __restrict__ bl)
{
    size_t i = (size_t)blockIdx.x * blockDim.x + threadIdx.x;
    float4 v = Y[i];
    float4 b = bl[threadIdx.x];
    v.x = fmaxf(v.x + b.x, 0.f);
    v.y = fmaxf(v.y + b.y, 0.f);
    v.z = fmaxf(v.z + b.z, 0.f);
    v.w = fmaxf(v.w + b.w, 0.f);
    Y[i] = v;
}

__global__ void zero_f(float* __restrict__ p, int n)
{
    int i = blockIdx.x * 256 + threadIdx.x;
    if (i < n) p[i] = 0.0f;
}

__global__ void count_deg(const int* __restrict__ ei, float* __restrict__ DEG)
{
    int e = blockIdx.x * 256 + threadIdx.x;
    if (e < Eg) atomicAdd(&DEG[ei[Eg + e]], 1.0f);
}

// d_out = [ x (Nn x 128) flat , attn (Bg x 256) flat ]   (float4)
__global__ void copy_out(const float4* __restrict__ X, const float4* __restrict__ G,
                         float4* __restrict__ out)
{
    size_t i = (size_t)blockIdx.x * 256 + threadIdx.x;
    const size_t nx = (size_t)Nn * 128 / 4;
    if (i < nx)                           out[i] = X[i];
    else if (i < nx + (size_t)Bg*Hh/4)    out[i] = G[i - nx];
}

// ---------------------------------------------------------------------------

static inline void launch_gemm(const float* A, int lda, const float* W, int ldw,
                               float* C, int ldc, int M, int Nc, int K, int accum,
                               hipStream_t s)
{
    dim3 grid(Nc / TN, M / TM);
    wmma_gemm_f32<<<grid, 256, 0, s>>>(A, lda, W, ldw, C, ldc, K, accum);
}

extern "C" void kernel_launch(void* const* d_in, const int* in_sizes, int n_in,
                              void* d_out, int out_size, void* d_ws, size_t ws_size,
                              hipStream_t stream)
{
    const float* x0 = (const float*)d_in[0];
    const int*   ei = (const int*)d_in[1];
    // d_in[2] = batch_sizes (constant 200, unused)

    // attn_params: 4 layers x (Wa, ba, Wf, bf, Wg, bg)
    const float* AP[4][6];
    int idx = 3;
    for (int i = 0; i < 4; ++i)
        for (int j = 0; j < 6; ++j) AP[i][j] = (const float*)d_in[idx++];
    // gnn_params: 3 layers x (Wl, bl, Wr)
    const float* GP[3][3];
    for (int i = 0; i < 3; ++i)
        for (int j = 0; j < 3; ++j) GP[i][j] = (const float*)d_in[idx++];

    // Workspace carve (floats)
    float* w    = (float*)d_ws;
    float* XA   = w;  w += (size_t)Nn * 256;
    float* XB   = w;  w += (size_t)Nn * 256;
    float* XCAT = w;  w += (size_t)Nn * 512;
    float* AGG  = w;  w += (size_t)Nn * 512;
    float* P    = w;  w += (size_t)Bg * 512;
    float* G    = w;  w += (size_t)Bg * 256;
    float* GT   = w;  w += (size_t)Bg * 256;
    float* DEG  = w;  w += (size_t)Nn;
    float* T1   = XCAT;   // attention logits alias (free during attention phase)
    float* T2   = AGG;    // attention features alias

    // init: attn state g = 0, degrees
    zero_f<<<(Bg * 256 + 255) / 256, 256, 0, stream>>>(G, Bg * 256);
    zero_f<<<(Nn + 255) / 256, 256, 0, stream>>>(DEG, Nn);
    count_deg<<<(Eg + 255) / 256, 256, 0, stream>>>(ei, DEG);

    const int dIn[3]  = {128, 256, 256};
    const int dOut[3] = {256, 256, 128};
    const float* xcur = x0;
    float* xbuf[2] = {XA, XB};

    for (int L = 0; L < 3; ++L) {
        const int d  = dIn[L];
        const int dc = d + Hh;
        const int dn = dOut[L];

        // ---- attention layer L ----
        launch_gemm(xcur, d, AP[L][0], Hh, T1, Hh, Nn, Hh, d, 0, stream);  // logits
        launch_gemm(xcur, d, AP[L][2], Hh, T2, Hh, Nn, Hh, d, 0, stream);  // feat
        pool_attn<<<Bg, 256, 0, stream>>>(T1, T2, AP[L][1], AP[L][3], G, P);
        launch_gemm(P, 512, AP[L][4], Hh, GT, Hh, Bg, Hh, 512, 0, stream); // [pooled||g]@Wg
        tanh_add<<<Bg, 256, 0, stream>>>(GT, AP[L][5], G);

        // ---- SAGE layer L ----
        prep_sage<<<Nn, 256, 0, stream>>>(xcur, G, DEG, XCAT, AGG, d);
        scatter_add<<<Eg, d, 0, stream>>>(ei, xcur, AGG, d, dc);
        float* Y = xbuf[L & 1];
        launch_gemm(AGG,  dc, GP[L][0], dn, Y, dn, Nn, dn, dc, 0, stream); // agg@Wl
        launch_gemm(XCAT, dc, GP[L][2], dn, Y, dn, Nn, dn, dc, 1, stream); // += xcat@Wr
        bias_relu<<<Nn, dn / 4, 0, stream>>>((float4*)Y, (const float4*)GP[L][1]);
        xcur = Y;
    }

    // ---- final attention layer (params[3], d=128) ----
    launch_gemm(xcur, 128, AP[3][0], Hh, T1, Hh, Nn, Hh, 128, 0, stream);
    launch_gemm(xcur, 128, AP[3][2], Hh, T2, Hh, Nn, Hh, 128, 0, stream);
    pool_attn<<<Bg, 256, 0, stream>>>(T1, T2, AP[3][1], AP[3][3], G, P);
    launch_gemm(P, 512, AP[3][4], Hh, GT, Hh, Bg, Hh, 512, 0, stream);
    tanh_add<<<Bg, 256, 0, stream>>>(GT, AP[3][5], G);

    // ---- outputs: x (Nn x 128) then attn (Bg x 256) ----
    const size_t total4 = ((size_t)Nn * 128 + (size_t)Bg * 256) / 4;
    copy_out<<<(unsigned)((total4 + 255) / 256), 256, 0, stream>>>(
        (const float4*)xcur, (const float4*)G, (float4*)d_out);
}